// EfficientPose_16698832846860
// MI455X (gfx1250) — compile-verified
//
#include <hip/hip_runtime.h>
#include <hip/hip_bf16.h>

// ---------------- problem constants ----------------
#define NN        150381
#define CC        90
#define PRE_K     1024
#define MAX_DET   100
#define CAND_MAX  4096          // expected survivors/class ~601 (thr=0.996); 4096 = huge headroom
#define SCORE_THR 0.996f
#define NMS_THR   0.5f
#define IMG_SZ    896.0f

// ---------------- workspace layout (bytes) ----------------
__host__ __device__ constexpr size_t alignup(size_t x) { return (x + 255) & ~size_t(255); }
constexpr size_t OFF_BOXES  = 0;                                               // N*4 f32 (decoded, clipped)
constexpr size_t OFF_TRANS  = alignup(OFF_BOXES  + size_t(NN) * 4 * 4);        // N*3 f32
constexpr size_t OFF_CCNT   = alignup(OFF_TRANS  + size_t(NN) * 3 * 4);        // C   i32
constexpr size_t OFF_CSCORE = alignup(OFF_CCNT   + size_t(CC) * 4);            // C*CAND_MAX f32
constexpr size_t OFF_CIDX   = alignup(OFF_CSCORE + size_t(CC) * CAND_MAX * 4); // C*CAND_MAX i32
constexpr size_t OFF_TKSC   = alignup(OFF_CIDX   + size_t(CC) * CAND_MAX * 4); // C*PRE_K f32 (sorted desc)
constexpr size_t OFF_TKAN   = alignup(OFF_TKSC   + size_t(CC) * PRE_K * 4);    // C*PRE_K i32 (anchor idx)
constexpr size_t OFF_TKBX   = alignup(OFF_TKAN   + size_t(CC) * PRE_K * 4);    // C*PRE_K*4 f32 (gathered boxes)
constexpr size_t OFF_KEPT   = alignup(OFF_TKBX   + size_t(CC) * PRE_K * 4 * 4);// C*PRE_K f32 (post-NMS scores)

// ------- TDM descriptor helpers (ISA cdna5 §8.3/8.4; amdgpu-toolchain 6-arg builtin) --------
typedef unsigned int v4u_t __attribute__((ext_vector_type(4)));
typedef int          v8i_t __attribute__((ext_vector_type(8)));
typedef int          v4i_t __attribute__((ext_vector_type(4)));

__device__ __forceinline__ v4u_t tdm_group0(unsigned lds_byte_addr, unsigned long long gaddr) {
  v4u_t g;
  g[0] = 1u;                                   // count=1 valid descriptor, user mode, no gather
  g[1] = lds_byte_addr;                        // lds_addr[31:0]
  g[2] = (unsigned)(gaddr & 0xFFFFFFFFull);    // global_addr[31:0]
  g[3] = (unsigned)((gaddr >> 32) & 0x01FFFFFFull) | (2u << 30); // global_addr[56:32] | type=2
  return g;
}
// 1-D tile of `nelem` 4-byte elements (nelem <= 65535 for tile_dim0)
__device__ __forceinline__ v8i_t tdm_group1_1d(unsigned nelem) {
  v8i_t g;
  g[0] = (int)(2u << 16);                        // workgroup_mask=0 (not in cluster), data_size=2 (4B)
  g[1] = (int)((nelem & 0xFFFFu) << 16);         // tensor_dim0[15:0]
  g[2] = (int)(((nelem >> 16) & 0xFFFFu) | (1u << 16)); // tensor_dim0[31:16], tensor_dim1=1
  g[3] = (int)((nelem & 0xFFFFu) << 16);         // tile_dim0
  g[4] = 0;                                      // tile_dim1=0 (unused), tile_dim2=0
  g[5] = (int)nelem;                             // tensor_dim0_stride[31:0]
  g[6] = 0;                                      // stride0[47:32]=0, tensor_dim1_stride lo = 0
  g[7] = 0;
  return g;
}

// ================= K1: decode boxes + translation, zero per-class counters =================
__global__ __launch_bounds__(256) void k_decode(
    const float* __restrict__ anchors, const float* __restrict__ regression,
    const float* __restrict__ tanch,   const float* __restrict__ tdelt,
    float* __restrict__ boxes, float* __restrict__ trans, int* __restrict__ ccnt) {
  int i = blockIdx.x * blockDim.x + threadIdx.x;
  if (i < CC) ccnt[i] = 0;
  if (i >= NN) return;
  float ax1 = anchors[i * 4 + 0], ay1 = anchors[i * 4 + 1];
  float ax2 = anchors[i * 4 + 2], ay2 = anchors[i * 4 + 3];
  float cxa = (ax1 + ax2) * 0.5f, cya = (ay1 + ay2) * 0.5f;
  float wa = ax2 - ax1, ha = ay2 - ay1;
  float ty = regression[i * 4 + 0], tx = regression[i * 4 + 1];
  float th = regression[i * 4 + 2], tw = regression[i * 4 + 3];
  float w = expf(tw) * wa, h = expf(th) * ha;
  float cy = ty * ha + cya, cx = tx * wa + cxa;
  boxes[i * 4 + 0] = fminf(fmaxf(cx - w * 0.5f, 0.0f), IMG_SZ);
  boxes[i * 4 + 1] = fminf(fmaxf(cy - h * 0.5f, 0.0f), IMG_SZ);
  boxes[i * 4 + 2] = fminf(fmaxf(cx + w * 0.5f, 0.0f), IMG_SZ);
  boxes[i * 4 + 3] = fminf(fmaxf(cy + h * 0.5f, 0.0f), IMG_SZ);
  float stride = tanch[i * 3 + 2];
  trans[i * 3 + 0] = tanch[i * 3 + 0] + tdelt[i * 3 + 0] * stride;
  trans[i * 3 + 1] = tanch[i * 3 + 1] + tdelt[i * 3 + 1] * stride;
  trans[i * 3 + 2] = tdelt[i * 3 + 2];
}

// ================= K2: streaming threshold scan of classification (54 MB, memory-bound) ======
__global__ __launch_bounds__(256) void k_collect(
    const float* __restrict__ cls, int* __restrict__ ccnt,
    float* __restrict__ cscore, int* __restrict__ cidx) {
  int g = blockIdx.x * blockDim.x + threadIdx.x;
  const int TOT = NN * CC;
  if (g >= TOT) return;
  if (((g & 2047) == 0) && (g + 16384 < TOT))
    __builtin_prefetch(cls + g + 16384, 0, 1);   // -> global_prefetch_b8
  float v = cls[g];
  if (v > SCORE_THR) {
    int c = g % CC;
    int a = g / CC;
    int pos = atomicAdd(&ccnt[c], 1);
    if (pos < CAND_MAX) {
      cscore[c * CAND_MAX + pos] = v;
      cidx[c * CAND_MAX + pos] = a;
    }
  }
}

// ================= K3: per-class bitonic sort (score desc, idx asc) -> top PRE_K =============
__global__ __launch_bounds__(1024) void k_topk(
    const int* __restrict__ ccnt, const float* __restrict__ cscore, const int* __restrict__ cidx,
    const float* __restrict__ boxes,
    float* __restrict__ tksc, int* __restrict__ tkan, float* __restrict__ tkbx) {
  __shared__ float key[CAND_MAX];
  __shared__ int   pay[CAND_MAX];
  const int c = blockIdx.x, t = threadIdx.x;
  int n = ccnt[c];
  if (n > CAND_MAX) n = CAND_MAX;
  for (int s = t; s < CAND_MAX; s += 1024) {
    if (s < n) { key[s] = cscore[c * CAND_MAX + s]; pay[s] = cidx[c * CAND_MAX + s]; }
    else       { key[s] = -1.0f;                    pay[s] = 0x40000000 + s; } // distinct pad ids
  }
  __syncthreads();
  for (int k = 2; k <= CAND_MAX; k <<= 1) {
    for (int j = k >> 1; j > 0; j >>= 1) {
      for (int s = t; s < CAND_MAX; s += 1024) {
        int l = s ^ j;
        if (l > s) {
          float ks = key[s], kl = key[l];
          int   ps = pay[s], pl = pay[l];
          // bef: element s precedes element l in (score desc, idx asc) order
          bool bef = (ks > kl) || (ks == kl && ps < pl);
          bool doSwap = ((s & k) == 0) ? (!bef) : bef;
          if (doSwap) { key[s] = kl; key[l] = ks; pay[s] = pl; pay[l] = ps; }
        }
      }
      __syncthreads();
    }
  }
  // t in [0,1024) == PRE_K slots
  float v = key[t];
  int a = pay[t];
  bool valid = v > SCORE_THR;
  tksc[c * PRE_K + t] = v;
  tkan[c * PRE_K + t] = valid ? a : 0;
  float b0 = 0.f, b1 = 0.f, b2 = 0.f, b3 = 0.f;
  if (valid) { b0 = boxes[a*4+0]; b1 = boxes[a*4+1]; b2 = boxes[a*4+2]; b3 = boxes[a*4+3]; }
  tkbx[(c * PRE_K + t) * 4 + 0] = b0;
  tkbx[(c * PRE_K + t) * 4 + 1] = b1;
  tkbx[(c * PRE_K + t) * 4 + 2] = b2;
  tkbx[(c * PRE_K + t) * 4 + 3] = b3;
}

// ================= K4: per-class greedy NMS; tiles staged in LDS via TDM ====================
__global__ __launch_bounds__(1024) void k_nms(
    const float* __restrict__ tkbx, const float* __restrict__ tksc,
    float* __restrict__ kept) {
  __shared__ float bx[PRE_K * 4];   // 16 KB, TDM destination
  __shared__ float sc[PRE_K];       //  4 KB, TDM destination
  __shared__ int   keep[PRE_K];
  const int c = blockIdx.x, t = threadIdx.x;

  if (t == 0) {
    // Issue the two tile DMAs from one wave (TDM ignores EXEC; other waves branch around).
    unsigned lds_bx = (unsigned)(size_t)(void*)&bx[0];
    unsigned lds_sc = (unsigned)(size_t)(void*)&sc[0];
    unsigned long long ga_bx = (unsigned long long)(size_t)(const void*)(tkbx + (size_t)c * PRE_K * 4);
    unsigned long long ga_sc = (unsigned long long)(size_t)(const void*)(tksc + (size_t)c * PRE_K);
    v4i_t z4; z4[0] = 0; z4[1] = 0; z4[2] = 0; z4[3] = 0;   // groups 2/3 unused (<=2D tensor)
    v8i_t z8; z8[0] = 0; z8[1] = 0; z8[2] = 0; z8[3] = 0;
    z8[4] = 0; z8[5] = 0; z8[6] = 0; z8[7] = 0;             // extra group (clang-23 6-arg form)
    __builtin_amdgcn_tensor_load_to_lds(tdm_group0(lds_bx, ga_bx), tdm_group1_1d(PRE_K * 4),
                                        z4, z4, z8, 0);
    __builtin_amdgcn_tensor_load_to_lds(tdm_group0(lds_sc, ga_sc), tdm_group1_1d(PRE_K),
                                        z4, z4, z8, 0);
    __builtin_amdgcn_s_wait_tensorcnt(0);
  }
  __syncthreads();

  float x1 = bx[t * 4 + 0], y1 = bx[t * 4 + 1], x2 = bx[t * 4 + 2], y2 = bx[t * 4 + 3];
  float areaT = (x2 - x1) * (y2 - y1);
  float s = sc[t];
  keep[t] = (s > SCORE_THR) ? 1 : 0;
  __syncthreads();

  for (int i = 0; i < PRE_K; ++i) {
    if (sc[i] <= SCORE_THR) break;           // sorted desc -> uniform exit for whole block
    // iteration i: reads keep[i]; writes only keep[t] with t>i  => race-free with one barrier
    if (keep[i] && t > i && keep[t]) {
      float bx1 = bx[i*4+0], by1 = bx[i*4+1], bx2 = bx[i*4+2], by2 = bx[i*4+3];
      float ai  = (bx2 - bx1) * (by2 - by1);
      float iw  = fmaxf(fminf(x2, bx2) - fmaxf(x1, bx1), 0.0f);
      float ih  = fmaxf(fminf(y2, by2) - fmaxf(y1, by1), 0.0f);
      float inter = iw * ih;
      float uni = fmaxf(areaT + ai - inter, 1e-9f);
      if (inter / uni > NMS_THR) keep[t] = 0;
    }
    __syncthreads();
  }
  kept[c * PRE_K + t] = keep[t] ? s : -1.0f;
}

// ================= K5: global top-100 (tie-break lowest flat idx) + gather ==================
__global__ __launch_bounds__(1024) void k_final(
    float* __restrict__ kept, const int* __restrict__ tkan,
    const float* __restrict__ boxes, const float* __restrict__ rotation,
    const float* __restrict__ trans, float* __restrict__ out) {
  __shared__ float bs[1024];
  __shared__ int   bi[1024];
  __shared__ float rs[MAX_DET];
  __shared__ int   rf[MAX_DET];
  const int t = threadIdx.x;
  const int TOT = CC * PRE_K;
  for (int d = 0; d < MAX_DET; ++d) {
    float best = -1e30f; int bidx = TOT;
    for (int s = t; s < TOT; s += 1024) {
      float v = kept[s];
      if (v > best) { best = v; bidx = s; }        // strict > keeps lowest index on ties
    }
    bs[t] = best; bi[t] = bidx;
    __syncthreads();
    for (int off = 512; off > 0; off >>= 1) {
      if (t < off) {
        float vo = bs[t + off]; int io = bi[t + off];
        if (vo > bs[t] || (vo == bs[t] && io < bi[t])) { bs[t] = vo; bi[t] = io; }
      }
      __syncthreads();
    }
    if (t == 0) { rs[d] = bs[0]; rf[d] = bi[0]; kept[bi[0]] = -1e30f; } // kept[] rebuilt by K4 each call
    __syncthreads();
  }
  if (t < MAX_DET) {
    float s = rs[t];
    int flat = rf[t];
    bool ok = s > SCORE_THR;
    int cls = flat >> 10;                // / PRE_K
    int a = tkan[flat];
    for (int j = 0; j < 4; ++j) out[t * 4 + j]       = ok ? boxes[a * 4 + j]    : -1.0f;
    out[400 + t]                                     = ok ? s                   : -1.0f;
    out[500 + t]                                     = ok ? (float)cls          : -1.0f; // labels as float
    for (int j = 0; j < 3; ++j) out[600 + t * 3 + j] = ok ? rotation[a * 3 + j] : -1.0f;
    for (int j = 0; j < 3; ++j) out[900 + t * 3 + j] = ok ? trans[a * 3 + j]    : -1.0f;
  }
}

// =============================== launch ===============================
extern "C" void kernel_launch(void* const* d_in, const int* in_sizes, int n_in,
                              void* d_out, int out_size, void* d_ws, size_t ws_size,
                              hipStream_t stream) {
  (void)in_sizes; (void)n_in; (void)out_size; (void)ws_size;
  const float* anchors    = (const float*)d_in[0];
  const float* regression = (const float*)d_in[1];
  const float* cls        = (const float*)d_in[2];
  const float* rotation   = (const float*)d_in[3];
  const float* tanch      = (const float*)d_in[4];
  const float* tdelt      = (const float*)d_in[5];
  float* out = (float*)d_out;
  char*  ws  = (char*)d_ws;

  float* boxes  = (float*)(ws + OFF_BOXES);
  float* trans  = (float*)(ws + OFF_TRANS);
  int*   ccnt   = (int*)  (ws + OFF_CCNT);
  float* cscore = (float*)(ws + OFF_CSCORE);
  int*   cidx   = (int*)  (ws + OFF_CIDX);
  float* tksc   = (float*)(ws + OFF_TKSC);
  int*   tkan   = (int*)  (ws + OFF_TKAN);
  float* tkbx   = (float*)(ws + OFF_TKBX);
  float* kept   = (float*)(ws + OFF_KEPT);

  k_decode<<<(NN + 255) / 256, 256, 0, stream>>>(anchors, regression, tanch, tdelt,
                                                 boxes, trans, ccnt);
  k_collect<<<(NN * CC + 255) / 256, 256, 0, stream>>>(cls, ccnt, cscore, cidx);
  k_topk<<<CC, 1024, 0, stream>>>(ccnt, cscore, cidx, boxes, tksc, tkan, tkbx);
  k_nms<<<CC, 1024, 0, stream>>>(tkbx, tksc, kept);
  k_final<<<1, 1024, 0, stream>>>(kept, tkan, boxes, rotation, trans, out);
}